// GaussianMaskInpainting_58025008169111
// MI455X (gfx1250) — compile-verified
//
#include <hip/hip_runtime.h>
#include <stdint.h>

// Gaussian mask inpainting, MI455X (gfx1250).
// mask depends only on (t, d2) with d2 = (x-cx)^2+(y-cy)^2 an integer in [0,1922].
// Kernel 1 builds a 50x1923 cumulative-product table (96K exps total).
// Kernel 2 streams x0 -> out at HBM roofline: per block (one batch element) the
// table row is DMA'd into LDS with the Tensor Data Mover (tensor_load_to_lds +
// s_wait_tensorcnt), per-pixel masks are gathered from LDS, and 3 channels are
// applied with coalesced float4 (b128) traffic.

#define S_IMG   32
#define NPIX    1024            // 32*32
#define ND2     1923            // 0 .. 31^2+31^2 = 1922
#define ND2_PAD 1936            // 16B-friendly padding
#define T_MAX   50
#define THREADS 256

typedef __attribute__((ext_vector_type(4))) unsigned int tdm_u32x4;
typedef __attribute__((ext_vector_type(8))) int          tdm_i32x8;
typedef __attribute__((ext_vector_type(4))) int          tdm_i32x4;

// ---------------------------------------------------------------------------
// Kernel 1: table[tt][d2] = prod_{i=1..tt+1} (1 - exp(-d2 / (2*(1+0.1*(i-1)))))
// ---------------------------------------------------------------------------
__global__ void gmi_build_table(float* __restrict__ table) {
    int d2 = blockIdx.x * THREADS + threadIdx.x;
    if (d2 >= ND2) return;
    float fd2 = (float)d2;
    float acc = 1.0f;
#pragma unroll 1
    for (int i = 1; i <= T_MAX; ++i) {
        float beta = __fadd_rn(1.0f, __fmul_rn(0.1f, (float)(i - 1)));
        float g    = expf(__fdiv_rn(-fd2, __fmul_rn(2.0f, beta)));
        acc        = __fmul_rn(acc, __fsub_rn(1.0f, g));
        table[(size_t)(i - 1) * ND2 + d2] = acc;
    }
}

// ---------------------------------------------------------------------------
// Shared apply stage: out[b,c,p] = x0[b,c,p] * lmask[p], float4 coalesced.
// 3072 floats per batch = 768 float4; 256 threads -> 3 float4 each.
// ---------------------------------------------------------------------------
__device__ __forceinline__ void gmi_apply(const float* __restrict__ x0,
                                          float* __restrict__ out,
                                          const float* lmask,
                                          int b, int tid) {
    const float4* __restrict__ xin  = (const float4*)(x0 + (size_t)b * 3 * NPIX);
    float4* __restrict__       xout = (float4*)(out + (size_t)b * 3 * NPIX);
    const float4* lm4 = (const float4*)lmask;
#pragma unroll
    for (int v = 0; v < 3; ++v) {
        int f = v * THREADS + tid;               // float4 index in [0,768)
        float4 a = xin[f];
        float4 m = lm4[f & (NPIX / 4 - 1)];      // f % 256 (channel-periodic)
        a.x = __fmul_rn(a.x, m.x);
        a.y = __fmul_rn(a.y, m.y);
        a.z = __fmul_rn(a.z, m.z);
        a.w = __fmul_rn(a.w, m.w);
        xout[f] = a;
    }
}

// ---------------------------------------------------------------------------
// Kernel 2 (table path): one block per batch element.
// ---------------------------------------------------------------------------
__global__ void gmi_apply_table(const float* __restrict__ x0,
                                const int*   __restrict__ t,
                                const float* __restrict__ cx,
                                const float* __restrict__ cy,
                                const float* __restrict__ table,
                                float* __restrict__ out) {
    __shared__ __align__(16) float ltab[ND2_PAD];
    __shared__ __align__(16) float lmask[NPIX];

    const int b   = blockIdx.x;
    const int tid = threadIdx.x;

    const int tb = __builtin_amdgcn_readfirstlane(t[b]);   // uniform, 1..50
    const float* row = table + (size_t)(tb - 1) * ND2;

#if defined(__HIP_DEVICE_COMPILE__) && __has_builtin(__builtin_amdgcn_tensor_load_to_lds)
    if (tid < 32) {  // wave 0 issues one TDM DMA: 1923 f32 row -> LDS
        uint64_t ga = (uint64_t)(uintptr_t)row;
        uint32_t la = (uint32_t)(uintptr_t)(void*)ltab;    // low 32 bits = LDS byte offset
        tdm_u32x4 g0;
        g0[0] = 1u;                                        // count=1, no gather
        g0[1] = la;                                        // lds_addr
        g0[2] = (uint32_t)ga;                              // global_addr[31:0]
        g0[3] = (uint32_t)((ga >> 32) & 0x1FFFFFFu)        // global_addr[56:32]
              | (2u << 30);                                // type = 2 ("image")
        tdm_i32x8 g1;
        g1[0] = (int)(2u << 16);                           // workgroup_mask=0, data_size=4B
        g1[1] = (int)((unsigned)ND2 << 16);                // tensor_dim0[15:0]
        g1[2] = (int)(1u << 16);                           // tensor_dim1 = 1
        g1[3] = (int)((unsigned)ND2 << 16);                // tile_dim0 = 1923
        g1[4] = 1;                                         // tile_dim1 = 1, tile_dim2 = 0
        g1[5] = ND2;                                       // tensor_dim0_stride
        g1[6] = 0;
        g1[7] = 0;
        tdm_i32x4 gz4;  gz4[0] = 0; gz4[1] = 0; gz4[2] = 0; gz4[3] = 0;
        tdm_i32x8 gz8;  gz8[0] = 0; gz8[1] = 0; gz8[2] = 0; gz8[3] = 0;
                        gz8[4] = 0; gz8[5] = 0; gz8[6] = 0; gz8[7] = 0;
        // amdgpu-toolchain (clang-23 / therock-10.0): 6-arg form
        __builtin_amdgcn_tensor_load_to_lds(g0, g1, gz4, gz4, gz8, 0);
#if __has_builtin(__builtin_amdgcn_s_wait_tensorcnt)
        __builtin_amdgcn_s_wait_tensorcnt(0);
#else
        asm volatile("s_wait_tensorcnt 0x0" ::: "memory");
#endif
    }
#else
    for (int i = tid; i < ND2; i += THREADS) ltab[i] = row[i];
#endif

    const int icx = (int)cx[b];
    const int icy = (int)cy[b];
    __syncthreads();

    // per-pixel mask gather from LDS table row
#pragma unroll
    for (int k = 0; k < NPIX / THREADS; ++k) {
        int p  = k * THREADS + tid;
        int dx = (p & 31) - icx;
        int dy = (p >> 5) - icy;
        lmask[p] = ltab[dx * dx + dy * dy];
    }
    __syncthreads();

    gmi_apply(x0, out, lmask, b, tid);
}

// ---------------------------------------------------------------------------
// Fallback (no workspace): compute mask directly with the exp loop.
// ---------------------------------------------------------------------------
__global__ void gmi_direct(const float* __restrict__ x0,
                           const int*   __restrict__ t,
                           const float* __restrict__ cx,
                           const float* __restrict__ cy,
                           float* __restrict__ out) {
    __shared__ __align__(16) float lmask[NPIX];
    const int b   = blockIdx.x;
    const int tid = threadIdx.x;
    const int tb  = t[b];
    const int icx = (int)cx[b];
    const int icy = (int)cy[b];

#pragma unroll
    for (int k = 0; k < NPIX / THREADS; ++k) {
        int p  = k * THREADS + tid;
        int dx = (p & 31) - icx;
        int dy = (p >> 5) - icy;
        float d2  = (float)(dx * dx + dy * dy);
        float acc = 1.0f;
#pragma unroll 1
        for (int i = 1; i <= tb; ++i) {
            float beta = __fadd_rn(1.0f, __fmul_rn(0.1f, (float)(i - 1)));
            float g    = expf(__fdiv_rn(-d2, __fmul_rn(2.0f, beta)));
            acc        = __fmul_rn(acc, __fsub_rn(1.0f, g));
        }
        lmask[p] = acc;
    }
    __syncthreads();
    gmi_apply(x0, out, lmask, b, tid);
}

// ---------------------------------------------------------------------------
extern "C" void kernel_launch(void* const* d_in, const int* in_sizes, int n_in,
                              void* d_out, int out_size, void* d_ws, size_t ws_size,
                              hipStream_t stream) {
    const float* x0 = (const float*)d_in[0];
    const int*   t  = (const int*)d_in[1];
    const float* cx = (const float*)d_in[2];
    const float* cy = (const float*)d_in[3];
    float* out = (float*)d_out;
    const int B = in_sizes[1];  // t has one entry per batch element

    const size_t table_bytes = (size_t)T_MAX * ND2 * sizeof(float);
    if (d_ws != nullptr && ws_size >= table_bytes) {
        gmi_build_table<<<(ND2 + THREADS - 1) / THREADS, THREADS, 0, stream>>>((float*)d_ws);
        gmi_apply_table<<<B, THREADS, 0, stream>>>(x0, t, cx, cy, (const float*)d_ws, out);
    } else {
        gmi_direct<<<B, THREADS, 0, stream>>>(x0, t, cx, cy, out);
    }
}